// SMPL_layer_19593640805170
// MI455X (gfx1250) — compile-verified
//
#include <hip/hip_runtime.h>
#include <math.h>

typedef __attribute__((ext_vector_type(2))) float v2f;
typedef __attribute__((ext_vector_type(8))) float v8f;

#define V_   6890
#define J_   24
#define NB_  10
#define P_   207
#define B_   1024
#define N3_  (V_*3)     // 20670
#define KPAD 208
#define NT_  1292       // ceil(20670/16)

// ---------------- K1: Rodrigues + pose_feature ----------------
__global__ __launch_bounds__(256) void k_rodrigues(
    const float* __restrict__ pose, const float* __restrict__ gori,
    float* __restrict__ rot_out, float* __restrict__ pf)
{
    int t = blockIdx.x * 256 + threadIdx.x;
    if (t >= B_ * J_) return;
    int b = t / J_, j = t % J_;
    float rx, ry, rz;
    if (j == 0) { rx = gori[b*3+0]; ry = gori[b*3+1]; rz = gori[b*3+2]; }
    else { int o = b*69 + (j-1)*3; rx = pose[o]; ry = pose[o+1]; rz = pose[o+2]; }
    float ax = rx + 1e-8f, ay = ry + 1e-8f, az = rz + 1e-8f;
    float ang = sqrtf(ax*ax + ay*ay + az*az);
    float inv = 1.0f / ang;
    float ux = rx*inv, uy = ry*inv, uz = rz*inv;
    float s = sinf(ang), c = cosf(ang);
    float K[9] = { 0.f, -uz,  uy,
                   uz,  0.f, -ux,
                  -uy,  ux,  0.f };
    float K2[9];
#pragma unroll
    for (int i = 0; i < 3; ++i)
#pragma unroll
        for (int q = 0; q < 3; ++q)
            K2[i*3+q] = K[i*3+0]*K[0*3+q] + K[i*3+1]*K[1*3+q] + K[i*3+2]*K[2*3+q];
    float R[9];
#pragma unroll
    for (int i = 0; i < 9; ++i) {
        float I = (i == 0 || i == 4 || i == 8) ? 1.0f : 0.0f;
        R[i] = I + s*K[i] + (1.0f - c)*K2[i];
    }
    float* ro = rot_out + (size_t)(b*J_ + j)*9;
#pragma unroll
    for (int i = 0; i < 9; ++i) ro[i] = R[i];
    if (j > 0) {
        float* pfo = pf + (size_t)b*KPAD + (j-1)*9;
#pragma unroll
        for (int i = 0; i < 9; ++i) {
            float I = (i == 0 || i == 4 || i == 8) ? 1.0f : 0.0f;
            pfo[i] = R[i] - I;
        }
    } else {
        pf[(size_t)b*KPAD + P_] = 0.0f;   // zero the K-pad element
    }
}

// ---------------- K2: Jrest = J_regressor @ v_shaped (v_shaped on the fly) ----
__global__ __launch_bounds__(256) void k_jrest(
    const float* __restrict__ betas, const float* __restrict__ vt,
    const float* __restrict__ sd, const float* __restrict__ Jr,
    float* __restrict__ Jrest)
{
    __shared__ float red[8*72];
    int b = blockIdx.x, tid = threadIdx.x;
    float bet[NB_];
#pragma unroll
    for (int l = 0; l < NB_; ++l) bet[l] = betas[b*NB_ + l];
    float acc[72];
#pragma unroll
    for (int i = 0; i < 72; ++i) acc[i] = 0.0f;

    for (int v = tid; v < V_; v += 256) {
        float vs[3];
#pragma unroll
        for (int k = 0; k < 3; ++k) {
            int n = v*3 + k;
            float x = vt[n];
            const float* sp = sd + (size_t)n*NB_;
#pragma unroll
            for (int l = 0; l < NB_; ++l) x += bet[l]*sp[l];
            vs[k] = x;
        }
#pragma unroll
        for (int j = 0; j < J_; ++j) {
            float w = Jr[(size_t)j*V_ + v];
            acc[j*3+0] += w*vs[0];
            acc[j*3+1] += w*vs[1];
            acc[j*3+2] += w*vs[2];
        }
    }
    int lane = tid & 31, wv = tid >> 5;
#pragma unroll
    for (int i = 0; i < 72; ++i) {
        float x = acc[i];
        x += __shfl_down(x, 16, 32);
        x += __shfl_down(x,  8, 32);
        x += __shfl_down(x,  4, 32);
        x += __shfl_down(x,  2, 32);
        x += __shfl_down(x,  1, 32);
        if (lane == 0) red[wv*72 + i] = x;
    }
    __syncthreads();
    if (tid < 72) {
        float s = 0.f;
#pragma unroll
        for (int w = 0; w < 8; ++w) s += red[w*72 + tid];
        Jrest[(size_t)b*72 + tid] = s;
    }
}

// ---------------- K3: v_posed = v_shaped + pose_feature @ posedirs (WMMA) -----
// Each wave: 64 rows (4 x 16x16 C tiles) x 16 cols; one B panel load feeds 4 WMMAs.
__global__ __launch_bounds__(128) void k_posegemm(
    const float* __restrict__ pf, const float* __restrict__ pd,
    const float* __restrict__ betas, const float* __restrict__ vt,
    const float* __restrict__ sd, float* __restrict__ vp)
{
    int wv = threadIdx.x >> 5, lane = threadIdx.x & 31;
    int tile = blockIdx.x*4 + wv;          // 16 M-blocks * 1292 N-tiles
    int mb = tile / NT_, nt = tile % NT_;
    int m0 = mb*64, n0 = nt*16;
    int half = lane >> 4, li = lane & 15;
    int col = n0 + li;
    int colc = (col < N3_) ? col : (N3_ - 1);

    const float* ap0 = pf + (size_t)(m0 +  0 + li)*KPAD;
    const float* ap1 = pf + (size_t)(m0 + 16 + li)*KPAD;
    const float* ap2 = pf + (size_t)(m0 + 32 + li)*KPAD;
    const float* ap3 = pf + (size_t)(m0 + 48 + li)*KPAD;

    v8f acc0 = {}, acc1 = {}, acc2 = {}, acc3 = {};

    // main loop: ka+1 <= 203 < 207, no bounds checks
    for (int kk = 0; kk < 204; kk += 4) {
        int ka = kk + 2*half;
        v2f bm;
        bm.x = pd[(size_t)ka*N3_ + colc];
        bm.y = pd[(size_t)(ka + 1)*N3_ + colc];
        v2f a0; a0.x = ap0[ka]; a0.y = ap0[ka + 1];
        v2f a1; a1.x = ap1[ka]; a1.y = ap1[ka + 1];
        v2f a2; a2.x = ap2[ka]; a2.y = ap2[ka + 1];
        v2f a3; a3.x = ap3[ka]; a3.y = ap3[ka + 1];
        acc0 = __builtin_amdgcn_wmma_f32_16x16x4_f32(false, a0, false, bm, (short)0, acc0, false, false);
        acc1 = __builtin_amdgcn_wmma_f32_16x16x4_f32(false, a1, false, bm, (short)0, acc1, false, false);
        acc2 = __builtin_amdgcn_wmma_f32_16x16x4_f32(false, a2, false, bm, (short)0, acc2, false, false);
        acc3 = __builtin_amdgcn_wmma_f32_16x16x4_f32(false, a3, false, bm, (short)0, acc3, false, false);
    }
    {   // peeled tail: kk = 204 ; ka = 204/206, only ka+1==207 is out of range
        int ka = 204 + 2*half;
        v2f bm;
        bm.x = pd[(size_t)ka*N3_ + colc];
        bm.y = (ka + 1 < P_) ? pd[(size_t)(ka + 1)*N3_ + colc] : 0.0f;
        v2f a0; a0.x = ap0[ka]; a0.y = ap0[ka + 1];   // pf[..207] == 0
        v2f a1; a1.x = ap1[ka]; a1.y = ap1[ka + 1];
        v2f a2; a2.x = ap2[ka]; a2.y = ap2[ka + 1];
        v2f a3; a3.x = ap3[ka]; a3.y = ap3[ka + 1];
        acc0 = __builtin_amdgcn_wmma_f32_16x16x4_f32(false, a0, false, bm, (short)0, acc0, false, false);
        acc1 = __builtin_amdgcn_wmma_f32_16x16x4_f32(false, a1, false, bm, (short)0, acc1, false, false);
        acc2 = __builtin_amdgcn_wmma_f32_16x16x4_f32(false, a2, false, bm, (short)0, acc2, false, false);
        acc3 = __builtin_amdgcn_wmma_f32_16x16x4_f32(false, a3, false, bm, (short)0, acc3, false, false);
    }

    // fused epilogue: add v_shaped (recomputed) and store v_posed
    bool cok = (col < N3_);
    float sdl[NB_];
    const float* sp = sd + (size_t)colc*NB_;
#pragma unroll
    for (int l = 0; l < NB_; ++l) sdl[l] = sp[l];
    float vtv = vt[colc];
    v8f accs[4] = { acc0, acc1, acc2, acc3 };
#pragma unroll
    for (int r = 0; r < 4; ++r) {
#pragma unroll
        for (int q = 0; q < 8; ++q) {
            int m = m0 + r*16 + q + 8*half;
            float base = vtv;
            const float* bp = betas + (size_t)m*NB_;
#pragma unroll
            for (int l = 0; l < NB_; ++l) base += bp[l]*sdl[l];
            if (cok) vp[(size_t)m*N3_ + col] = accs[r][q] + base;
        }
    }
}

// ---------------- K4: kinematic chain, joints out, A matrices ----------------
__global__ __launch_bounds__(256) void k_chain(
    const float* __restrict__ rot, const float* __restrict__ Jrest,
    float* __restrict__ joints_out, float* __restrict__ Amat)
{
    int b = blockIdx.x*256 + threadIdx.x;
    if (b >= B_) return;
    const int par[J_] = {-1,0,0,0,1,2,3,4,5,6,7,8,9,9,9,12,13,14,16,17,18,19,20,21};
    float Jr[J_][3];
#pragma unroll
    for (int j = 0; j < J_; ++j)
#pragma unroll
        for (int k = 0; k < 3; ++k) Jr[j][k] = Jrest[(size_t)b*72 + j*3 + k];

    float G[J_][12];
    const float* R0 = rot + (size_t)b*J_*9;
#pragma unroll
    for (int i = 0; i < 3; ++i) {
        G[0][i*4+0] = R0[i*3+0]; G[0][i*4+1] = R0[i*3+1]; G[0][i*4+2] = R0[i*3+2];
        G[0][i*4+3] = Jr[0][i];
    }
    for (int j = 1; j < J_; ++j) {
        const float* Rj = rot + (size_t)b*J_*9 + j*9;
        int p = par[j];
        float rel[3];
#pragma unroll
        for (int k = 0; k < 3; ++k) rel[k] = Jr[j][k] - Jr[p][k];
#pragma unroll
        for (int i = 0; i < 3; ++i) {
#pragma unroll
            for (int c = 0; c < 3; ++c)
                G[j][i*4+c] = G[p][i*4+0]*Rj[0*3+c] + G[p][i*4+1]*Rj[1*3+c] + G[p][i*4+2]*Rj[2*3+c];
            G[j][i*4+3] = G[p][i*4+0]*rel[0] + G[p][i*4+1]*rel[1] + G[p][i*4+2]*rel[2] + G[p][i*4+3];
        }
    }
#pragma unroll
    for (int j = 0; j < J_; ++j)
#pragma unroll
        for (int i = 0; i < 3; ++i)
            joints_out[(size_t)b*72 + j*3 + i] = G[j][i*4+3] - G[0][i*4+3];
#pragma unroll
    for (int j = 0; j < J_; ++j) {
        float* Ao = Amat + (size_t)b*288 + j*12;
#pragma unroll
        for (int i = 0; i < 3; ++i) {
            float tt = G[j][i*4+3] - (G[j][i*4+0]*Jr[j][0] + G[j][i*4+1]*Jr[j][1] + G[j][i*4+2]*Jr[j][2]);
            Ao[i*4+0] = G[j][i*4+0]; Ao[i*4+1] = G[j][i*4+1]; Ao[i*4+2] = G[j][i*4+2];
            Ao[i*4+3] = tt;
        }
    }
}

// ---------------- K5: skinning (WMMA) : Tv = w @ A, verts out ----------------
__global__ __launch_bounds__(128) void k_skin(
    const float* __restrict__ w, const float* __restrict__ Amat,
    const float* __restrict__ vp, const float* __restrict__ Jrest,
    float* __restrict__ verts)
{
    __shared__ float ctile[4][16][16];
    int b = blockIdx.y;
    int wv = threadIdx.x >> 5, lane = threadIdx.x & 31;
    int half = lane >> 4, li = lane & 15;
    int v0 = blockIdx.x*64 + wv*16;
    int vrow = v0 + li;
    int vrc  = (vrow < V_) ? vrow : (V_ - 1);
    const float* Ab = Amat + (size_t)b*288;

    v8f acc = {};
#pragma unroll
    for (int kk = 0; kk < J_; kk += 4) {
        int ka = kk + 2*half;
        v2f a;  a.x = w[(size_t)vrc*J_ + ka];  a.y = w[(size_t)vrc*J_ + ka + 1];
        v2f bm;
        bm.x = (li < 12) ? Ab[ka*12 + li]       : 0.0f;
        bm.y = (li < 12) ? Ab[(ka + 1)*12 + li] : 0.0f;
        acc = __builtin_amdgcn_wmma_f32_16x16x4_f32(
                false, a, false, bm, (short)0, acc, false, false);
    }
#pragma unroll
    for (int r = 0; r < 8; ++r)
        ctile[wv][r + 8*half][li] = acc[r];
    __syncthreads();

    if (half == 0) {
        int v = v0 + li;
        if (v < V_) {
            float T[12];
#pragma unroll
            for (int c = 0; c < 12; ++c) T[c] = ctile[wv][li][c];
            const float* pv = vp + (size_t)b*N3_ + (size_t)v*3;
            float p0 = pv[0], p1 = pv[1], p2 = pv[2];
            float r0 = Jrest[(size_t)b*72 + 0];
            float r1 = Jrest[(size_t)b*72 + 1];
            float r2 = Jrest[(size_t)b*72 + 2];
            float o0 = T[0]*p0 + T[1]*p1 + T[2]*p2  + T[3]  - r0;
            float o1 = T[4]*p0 + T[5]*p1 + T[6]*p2  + T[7]  - r1;
            float o2 = T[8]*p0 + T[9]*p1 + T[10]*p2 + T[11] - r2;
            float* vo = verts + (size_t)b*N3_ + (size_t)v*3;
            vo[0] = o0; vo[1] = o1; vo[2] = o2;
        }
    }
}

// ---------------- K6: joints_from_verts (h36m), root-subtracted --------------
__global__ __launch_bounds__(256) void k_jfv(
    const float* __restrict__ Jh, const float* __restrict__ verts,
    const float* __restrict__ Jrest, float* __restrict__ out)
{
    __shared__ float red[8*51];
    __shared__ float fin[51];
    int b = blockIdx.x, tid = threadIdx.x;
    float r0 = Jrest[(size_t)b*72 + 0];
    float r1 = Jrest[(size_t)b*72 + 1];
    float r2 = Jrest[(size_t)b*72 + 2];
    float acc[51];
#pragma unroll
    for (int i = 0; i < 51; ++i) acc[i] = 0.0f;

    for (int v = tid; v < V_; v += 256) {
        const float* pv = verts + (size_t)b*N3_ + (size_t)v*3;
        float x = pv[0] + r0, y = pv[1] + r1, z = pv[2] + r2;  // un-root to match ref
#pragma unroll
        for (int i = 0; i < 17; ++i) {
            float wv_ = Jh[(size_t)i*V_ + v];
            acc[i*3+0] += wv_*x; acc[i*3+1] += wv_*y; acc[i*3+2] += wv_*z;
        }
    }
    int lane = tid & 31, wv = tid >> 5;
#pragma unroll
    for (int i = 0; i < 51; ++i) {
        float x = acc[i];
        x += __shfl_down(x, 16, 32);
        x += __shfl_down(x,  8, 32);
        x += __shfl_down(x,  4, 32);
        x += __shfl_down(x,  2, 32);
        x += __shfl_down(x,  1, 32);
        if (lane == 0) red[wv*51 + i] = x;
    }
    __syncthreads();
    if (tid < 51) {
        float s = 0.f;
#pragma unroll
        for (int w8 = 0; w8 < 8; ++w8) s += red[w8*51 + tid];
        fin[tid] = s;
    }
    __syncthreads();
    if (tid < 51)
        out[(size_t)b*51 + tid] = fin[tid] - fin[tid % 3];
}

// ---------------- launch --------------------------------------------------
extern "C" void kernel_launch(void* const* d_in, const int* in_sizes, int n_in,
                              void* d_out, int out_size, void* d_ws, size_t ws_size,
                              hipStream_t stream)
{
    const float* pose  = (const float*)d_in[0];
    const float* betas = (const float*)d_in[1];
    const float* gori  = (const float*)d_in[2];
    const float* vt    = (const float*)d_in[3];
    const float* sd    = (const float*)d_in[4];
    const float* pd    = (const float*)d_in[5];
    const float* Jr    = (const float*)d_in[6];
    const float* Jh    = (const float*)d_in[7];
    const float* lw    = (const float*)d_in[8];

    float* out      = (float*)d_out;
    float* o_verts  = out;                       // B*V*3 = 21,166,080
    float* o_joints = out + 21166080;            // B*24*3
    float* o_rot    = out + 21239808;            // B*24*9
    float* o_jfv    = out + 21460992;            // B*17*3

    float* ws       = (float*)d_ws;
    float* ws_pf    = ws;                        // B*208   = 212,992
    float* ws_Jrest = ws_pf + (size_t)B_*KPAD;   // B*72    =  73,728
    float* ws_A     = ws_Jrest + (size_t)B_*72;  // B*288   = 294,912

    k_rodrigues<<<(B_*J_ + 255)/256, 256, 0, stream>>>(pose, gori, o_rot, ws_pf);
    k_jrest<<<B_, 256, 0, stream>>>(betas, vt, sd, Jr, ws_Jrest);
    // 16 M-blocks (64 rows) * 1292 N-tiles = 20,672 wave-tiles; 4 waves/block
    k_posegemm<<<5168, 128, 0, stream>>>(ws_pf, pd, betas, vt, sd, o_verts /*v_posed in place*/);
    k_chain<<<(B_ + 255)/256, 256, 0, stream>>>(o_rot, ws_Jrest, o_joints, ws_A);
    k_skin<<<dim3(108, B_), 128, 0, stream>>>(lw, ws_A, o_verts, ws_Jrest, o_verts);
    k_jfv<<<B_, 256, 0, stream>>>(Jh, o_verts, ws_Jrest, o_jfv);
}